// GraphAutoEncoder_84189948936516
// MI455X (gfx1250) — compile-verified
//
#include <hip/hip_runtime.h>

typedef float v2f __attribute__((ext_vector_type(2)));
typedef float v8f __attribute__((ext_vector_type(8)));

// ---------------------------------------------------------------------------
// WMMA f32 GEMM with LDS staging:
//   C[M,N] = act(A[M,K] @ B[K,N] + bias[N]),  act = relu or identity
// Block = 256 threads (8 waves) -> 16x128 output strip; grid = (M/16, N/128).
// K-chunk = 32 staged in LDS; per chunk each wave runs 8x v_wmma_f32_16x16x4_f32.
// Requires: M%16==0, N%128==0, K%32==0 (true for all four GEMMs here).
// ---------------------------------------------------------------------------
#define KC 32
#define SA_STRIDE 34   // 16 rows of 32 floats, pad 2: conflict-free b64 reads
#define SB_STRIDE 136  // 32 rows of 128 floats, pad 8: 2*136 % 64 == 16

__global__ __launch_bounds__(256) void gemm_wmma_lds(
    const float* __restrict__ A, const float* __restrict__ B,
    const float* __restrict__ bias, float* __restrict__ C,
    int M, int K, int N, int do_relu)
{
    __shared__ float sA[16 * SA_STRIDE];   //  2176 B
    __shared__ float sB[KC * SB_STRIDE];   // 17408 B

    const int t    = threadIdx.x;
    const int wave = t >> 5;               // 0..7
    const int lane = t & 31;
    const int lh   = lane >> 4;            // K-pair select (0: k{0,1}, 1: k{2,3})
    const int lm   = lane & 15;

    const int m0 = blockIdx.x * 16;        // output row base
    const int nb = blockIdx.y * 128;       // output col base of block
    const int wcol = wave * 16 + lm;       // this lane's column within the strip

    // staging coordinates
    const int ar = t >> 4;                 // A stage: row 0..15
    const int ac = (t & 15) * 2;           // A stage: col pair 0..30

    v8f acc = {};

    for (int kc = 0; kc < K; kc += KC) {
        // ---- stage A tile [16 x 32] (coalesced float2 loads) ----
        {
            const float2 av = *(const float2*)&A[(size_t)(m0 + ar) * K + kc + ac];
            sA[ar * SA_STRIDE + ac]     = av.x;
            sA[ar * SA_STRIDE + ac + 1] = av.y;
        }
        // ---- stage B tile [32 x 128] (coalesced float4 loads) ----
#pragma unroll
        for (int p = 0; p < 4; ++p) {
            const int row = p * 8 + wave;              // 0..31, each exactly once
            const float4 bv = *(const float4*)&B[(size_t)(kc + row) * N + nb + lane * 4];
            *(float4*)&sB[row * SB_STRIDE + lane * 4] = bv;
        }
        __syncthreads();

        // ---- 8 WMMA steps over the chunk ----
#pragma unroll
        for (int k8 = 0; k8 < 8; ++k8) {
            const int ka = k8 * 4 + 2 * lh;
            v2f a = *(const v2f*)&sA[lm * SA_STRIDE + ka];     // 8B-aligned ds_load_b64
            v2f b;
            b.x = sB[(ka)     * SB_STRIDE + wcol];
            b.y = sB[(ka + 1) * SB_STRIDE + wcol];
            acc = __builtin_amdgcn_wmma_f32_16x16x4_f32(
                false, a, false, b, (short)0, acc, false, false);
        }
        __syncthreads();
    }

    // ---- epilogue: bias (+ relu), scattered stores ----
    const int n = nb + wcol;
    const float bs = bias[n];
#pragma unroll
    for (int i = 0; i < 8; ++i) {
        const int m = m0 + i + 8 * lh;     // C layout: VGPR i -> M = i (+8 for hi half)
        float v = acc[i] + bs;
        if (do_relu) v = fmaxf(v, 0.0f);
        C[(size_t)m * N + n] = v;
    }
}

// ---------------------------------------------------------------------------
// Helpers for scatter-mean aggregation
// ---------------------------------------------------------------------------
__global__ __launch_bounds__(256) void zero_f32v4(float4* __restrict__ p, long long n4)
{
    long long i = (long long)blockIdx.x * blockDim.x + threadIdx.x;
    if (i < n4) p[i] = make_float4(0.f, 0.f, 0.f, 0.f);
}

__global__ __launch_bounds__(256) void count_edges(
    const long long* __restrict__ ei, long long E, float* __restrict__ cnt)
{
    long long e = (long long)blockIdx.x * blockDim.x + threadIdx.x;
    if (e < E) atomicAdd(&cnt[ei[E + e]], 1.0f);
}

// one block per edge; blockDim.x == F (feature width)
__global__ void aggregate_edges(
    const long long* __restrict__ ei, long long E,
    const float* __restrict__ xt, float* __restrict__ sum, int F)
{
    const long long e = blockIdx.x;
    const int f = threadIdx.x;
    const long long s = ei[e];
    const long long d = ei[E + e];
    atomicAdd(&sum[d * (long long)F + f], xt[s * (long long)F + f]);
}

// out[i] = relu(sum[i] / max(cnt[node],1)) + xt[i]   (out may alias xt)
__global__ __launch_bounds__(256) void combine_mean_relu_res(
    const float* __restrict__ sum, const float* __restrict__ cnt,
    const float* __restrict__ xt, float* __restrict__ out,
    int F, long long total)
{
    long long i = (long long)blockIdx.x * blockDim.x + threadIdx.x;
    if (i >= total) return;
    const long long node = i / F;
    const float c = fmaxf(cnt[node], 1.0f);
    const float mean = sum[i] / c;
    out[i] = fmaxf(mean, 0.0f) + xt[i];
}

// ---------------------------------------------------------------------------
// Orchestration
// ---------------------------------------------------------------------------
extern "C" void kernel_launch(void* const* d_in, const int* in_sizes, int n_in,
                              void* d_out, int out_size, void* d_ws, size_t ws_size,
                              hipStream_t stream)
{
    constexpr int IN_DIM = 256, HID = 256, LAT = 128, DEC_HID = 512;

    const float*     x   = (const float*)d_in[0];
    const long long* ei  = (const long long*)d_in[1];   // [2, E] int64
    const float*     W1  = (const float*)d_in[2];
    const float*     b1  = (const float*)d_in[3];
    const float*     W2  = (const float*)d_in[4];
    const float*     b2  = (const float*)d_in[5];
    const float*     Wd1 = (const float*)d_in[6];
    const float*     bd1 = (const float*)d_in[7];
    const float*     Wd2 = (const float*)d_in[8];
    const float*     bd2 = (const float*)d_in[9];

    const long long NM = in_sizes[0] / IN_DIM;          // 50000 nodes
    const long long E  = in_sizes[1] / 2;               // 800000 edges

    // workspace layout (floats):
    //   [0,       NM*256) : xt1 / h  (region0)
    //   [NM*256,  NM*512) : sum      (region1)  -- regions 0+1 reused as hd [NM,512]
    //   [NM*512,  NM*640) : xt2      (region2)
    //   [NM*640,  NM*641) : cnt
    float* ws   = (float*)d_ws;
    float* xt1  = ws;
    float* sum  = ws + NM * 256;
    float* xt2  = ws + NM * 512;
    float* cnt  = ws + NM * 640;
    float* hd   = ws;                                   // overlays regions 0+1

    float* x_hat = (float*)d_out;                          // [NM, 256]
    float* z     = (float*)d_out + NM * (long long)IN_DIM; // [NM, 128]

    const int TPB = 256;
    auto blocks = [](long long n, int tpb) { return (unsigned)((n + tpb - 1) / tpb); };

    const unsigned MB = (unsigned)(NM / 16);            // 3125 row-blocks

    // ---------- GCN layer 1 ----------
    gemm_wmma_lds<<<dim3(MB, HID / 128), TPB, 0, stream>>>(
        x, W1, b1, xt1, (int)NM, IN_DIM, HID, 0);
    zero_f32v4<<<blocks(NM * HID / 4, TPB), TPB, 0, stream>>>((float4*)sum, NM * HID / 4);
    zero_f32v4<<<blocks(NM / 4, TPB), TPB, 0, stream>>>((float4*)cnt, NM / 4);
    count_edges<<<blocks(E, TPB), TPB, 0, stream>>>(ei, E, cnt);
    aggregate_edges<<<(unsigned)E, HID, 0, stream>>>(ei, E, xt1, sum, HID);
    combine_mean_relu_res<<<blocks(NM * HID, TPB), TPB, 0, stream>>>(
        sum, cnt, xt1, xt1 /* h in-place */, HID, NM * HID);

    // ---------- GCN layer 2 ----------
    gemm_wmma_lds<<<dim3(MB, LAT / 128), TPB, 0, stream>>>(
        xt1, W2, b2, xt2, (int)NM, HID, LAT, 0);
    zero_f32v4<<<blocks(NM * LAT / 4, TPB), TPB, 0, stream>>>((float4*)sum, NM * LAT / 4);
    aggregate_edges<<<(unsigned)E, LAT, 0, stream>>>(ei, E, xt2, sum, LAT);
    combine_mean_relu_res<<<blocks(NM * LAT, TPB), TPB, 0, stream>>>(
        sum, cnt, xt2, z /* straight into d_out */, LAT, NM * LAT);

    // ---------- decoder ----------
    gemm_wmma_lds<<<dim3(MB, DEC_HID / 128), TPB, 0, stream>>>(
        z, Wd1, bd1, hd, (int)NM, LAT, DEC_HID, 1);
    gemm_wmma_lds<<<dim3(MB, IN_DIM / 128), TPB, 0, stream>>>(
        hd, Wd2, bd2, x_hat, (int)NM, DEC_HID, IN_DIM, 0);
}